// OddLayerFirst_1821066133816
// MI455X (gfx1250) — compile-verified
//
#include <hip/hip_runtime.h>

// out[b, e] = inputs[b, v_idx[e]]  -- pure column gather, bandwidth bound.
// Floor: ~512MB NT writes + ~128MB reads (input fits in 192MB L2) @ 23.3 TB/s ~ 27us.

typedef float f32x4 __attribute__((ext_vector_type(4)));
typedef int   i32x4 __attribute__((ext_vector_type(4)));

#if defined(__has_builtin)
#if __has_builtin(__builtin_amdgcn_global_load_async_to_lds_b128)
#define GATHER_HAS_ASYNC 1
#endif
#endif
#ifndef GATHER_HAS_ASYNC
#define GATHER_HAS_ASYNC 0
#endif

#define THREADS 256
#define EPT 4                 // edges per thread -> one b128 store per row
#define EPB (THREADS * EPT)   // 1024 edges per block

typedef __attribute__((address_space(1))) i32x4* gptr_i32x4;
typedef __attribute__((address_space(3))) i32x4* lptr_i32x4;

__launch_bounds__(THREADS)
__global__ void gather_cols_kernel(const float* __restrict__ in,
                                   const int* __restrict__ vidx,
                                   float* __restrict__ out,
                                   int V, int E, int rowsPerBlock)
{
    const int tid = threadIdx.x;
    const int e0  = blockIdx.x * EPB + tid * EPT;
    const int rowStart = blockIdx.y * rowsPerBlock;

    i32x4 idx;
#if GATHER_HAS_ASYNC
    // Stage this block's 4KB index chunk LDS-side with the CDNA5 async copy
    // (tracked by ASYNCcnt), then pull each thread's 4 indices into registers.
    __shared__ int sidx[EPB];
    {
        // builtin signature: (v4i global* src, v4i local* dst, imm offset, imm cpol)
        gptr_i32x4 gsrc = (gptr_i32x4)(i32x4*)(const_cast<int*>(vidx) + e0);
        lptr_i32x4 ldst = (lptr_i32x4)(i32x4*)(&sidx[tid * EPT]);
        __builtin_amdgcn_global_load_async_to_lds_b128(gsrc, ldst, 0, 0);
    }
#if __has_builtin(__builtin_amdgcn_s_wait_asynccnt)
    __builtin_amdgcn_s_wait_asynccnt(0);
#else
    asm volatile("s_wait_asynccnt 0" ::: "memory");
#endif
    __syncthreads();
    idx = *(const i32x4*)(&sidx[tid * EPT]);
#else
    idx = *(const i32x4*)(vidx + e0);   // coalesced b128 index load, once per thread
#endif

    const float* __restrict__ src = in  + (size_t)rowStart * (size_t)V;
    float*       __restrict__ dst = out + (size_t)rowStart * (size_t)E + (size_t)e0;

    // Indices live in registers across all rows; gather reads stay RT (L2-resident
    // input, ~4x reuse), output stores are non-temporal so the 512MB stream does
    // not evict the 128MB input from the 192MB L2.
    #pragma unroll 4
    for (int r = 0; r < rowsPerBlock; ++r) {
        f32x4 v;
        v.x = src[idx.x];
        v.y = src[idx.y];
        v.z = src[idx.z];
        v.w = src[idx.w];
        __builtin_nontemporal_store(v, (f32x4*)dst);
        src += V;
        dst += E;
    }
}

extern "C" void kernel_launch(void* const* d_in, const int* in_sizes, int n_in,
                              void* d_out, int out_size, void* d_ws, size_t ws_size,
                              hipStream_t stream) {
    const float* inputs = (const float*)d_in[0];   // [B, V] fp32
    const int*   vidx   = (const int*)d_in[1];     // [E]    int32
    float*       out    = (float*)d_out;           // [B, E] fp32

    const int       E       = in_sizes[1];                       // 32768
    const long long inElems = (long long)in_sizes[0];            // B*V
    const int       B       = (int)((long long)out_size / E);    // 4096
    const int       V       = (int)(inElems / B);                // 8192

    int rowsPerBlock = 32;
    while (rowsPerBlock > 1 && (B % rowsPerBlock)) rowsPerBlock >>= 1;

    dim3 grid(E / EPB, B / rowsPerBlock);
    gather_cols_kernel<<<grid, THREADS, 0, stream>>>(inputs, vidx, out, V, E, rowsPerBlock);
}